// GIN_71296457113907
// MI455X (gfx1250) — compile-verified
//
#include <hip/hip_runtime.h>
#include <hip/hip_bf16.h>

#define NN 50000
#define EE 800000
#define HH 128
#define GG 128
#define CC 10
#define LL 3
#define MT 80            // rows per block = 5 WMMA sub-tiles; 50000 = 80*625
#define BN_EPS 1e-5f

typedef __attribute__((ext_vector_type(2))) float v2f;
typedef __attribute__((ext_vector_type(8))) float v8f;

// ---------------- edge aggregation: agg[dst] += x[src] * ec ----------------
// one wave per edge; lane handles 4 contiguous features (32*4 = 128)
__global__ __launch_bounds__(256) void gin_aggregate(
    const float* __restrict__ xin, const float* __restrict__ ec,
    const int* __restrict__ src, const int* __restrict__ dst,
    float* __restrict__ agg)
{
    int lane = threadIdx.x & 31;
    int e = blockIdx.x * 8 + (threadIdx.x >> 5);
    if (e >= EE) return;
    int s = src[e];
    int d = dst[e];
    float w = ec[e];
    float4 v = ((const float4*)(xin + (size_t)s * HH))[lane];
    float* a = agg + (size_t)d * HH + lane * 4;
    atomicAdd(a + 0, v.x * w);
    atomicAdd(a + 1, v.y * w);
    atomicAdd(a + 2, v.z * w);
    atomicAdd(a + 3, v.w * w);
}

// ---------------- GEMM1: h1 = (agg*nc + x) @ W1 + b1, accumulate BN stats ---
// block = 8 waves, owns 80 rows (5 sub-tiles); wave w owns cols [16w,16w+16)
// A strip staged in LDS (padded stride 132); B fragment reused across 5 tiles
__global__ __launch_bounds__(256) void gin_gemm1(
    const float* __restrict__ agg, const float* __restrict__ xin,
    const float* __restrict__ nc, const float* __restrict__ W,
    const float* __restrict__ bias, float* __restrict__ h1,
    float* __restrict__ stat_sum, float* __restrict__ stat_sq)
{
    __shared__ float ldsA[MT * 132];
    int tid = threadIdx.x;
    int row0 = blockIdx.x * MT;

    // stage A = agg*nc + x  (80 rows x 128)
    for (int i = tid; i < MT * 32; i += 256) {
        int r = i >> 5;
        int c = i & 31;
        int row = row0 + r;
        float4 av = ((const float4*)(agg + (size_t)row * HH))[c];
        float4 xv = ((const float4*)(xin + (size_t)row * HH))[c];
        float s = nc[row];
        float4 h;
        h.x = av.x * s + xv.x;
        h.y = av.y * s + xv.y;
        h.z = av.z * s + xv.z;
        h.w = av.w * s + xv.w;
        ((float4*)(ldsA + r * 132))[c] = h;
    }
    __syncthreads();

    int lane = tid & 31;
    int wv = tid >> 5;
    int m = lane & 15;          // A row within sub-tile
    int half = lane >> 4;       // selects K pair (0/1 vs 2/3)
    int col = wv * 16 + (lane & 15);

    v8f acc[5] = {};
    for (int k0 = 0; k0 < HH; k0 += 4) {
        int ka = k0 + half * 2;
        v2f b;
        b.x = W[(size_t)ka * HH + col];
        b.y = W[(size_t)(ka + 1) * HH + col];
#pragma unroll
        for (int t = 0; t < 5; ++t) {
            v2f a;
            a.x = ldsA[(m + 16 * t) * 132 + ka];
            a.y = ldsA[(m + 16 * t) * 132 + ka + 1];
            acc[t] = __builtin_amdgcn_wmma_f32_16x16x4_f32(
                false, a, false, b, (short)0, acc[t], false, false);
        }
    }

    float bb = bias[col];
    float s = 0.f, sq = 0.f;
    int mbase = half * 8;
#pragma unroll
    for (int t = 0; t < 5; ++t) {
#pragma unroll
        for (int i = 0; i < 8; ++i) {
            float v = acc[t][i] + bb;
            h1[(size_t)(row0 + 16 * t + mbase + i) * HH + col] = v;
            s += v;
            sq += v * v;
        }
    }
    atomicAdd(&stat_sum[col], s);
    atomicAdd(&stat_sq[col], sq);
}

// ---------------- BN finalize: fold mean/var/gamma/beta into scale/shift ----
__global__ void gin_bnstats(const float* __restrict__ ssum, const float* __restrict__ ssq,
                            const float* __restrict__ g1, const float* __restrict__ be1,
                            float* __restrict__ scale, float* __restrict__ shift)
{
    int j = threadIdx.x;
    float mu = ssum[j] * (1.0f / (float)NN);
    float var = ssq[j] * (1.0f / (float)NN) - mu * mu;
    float rs = rsqrtf(var + BN_EPS);
    float sc = rs * g1[j];
    scale[j] = sc;
    shift[j] = be1[j] - mu * sc;
}

// ---------------- GEMM2: xout = relu( relu(h1*scale+shift) @ W2 + b2 ) ------
__global__ __launch_bounds__(256) void gin_gemm2(
    const float* __restrict__ h1, const float* __restrict__ scale,
    const float* __restrict__ shift, const float* __restrict__ W,
    const float* __restrict__ bias, float* __restrict__ xout)
{
    __shared__ float ldsA[MT * 132];
    int tid = threadIdx.x;
    int row0 = blockIdx.x * MT;

    // stage A = relu(h1*scale + shift)
    for (int i = tid; i < MT * 32; i += 256) {
        int r = i >> 5;
        int c = i & 31;
        int row = row0 + r;
        float4 hv = ((const float4*)(h1 + (size_t)row * HH))[c];
        float4 sc = ((const float4*)scale)[c];
        float4 sh = ((const float4*)shift)[c];
        float4 h;
        h.x = fmaxf(hv.x * sc.x + sh.x, 0.f);
        h.y = fmaxf(hv.y * sc.y + sh.y, 0.f);
        h.z = fmaxf(hv.z * sc.z + sh.z, 0.f);
        h.w = fmaxf(hv.w * sc.w + sh.w, 0.f);
        ((float4*)(ldsA + r * 132))[c] = h;
    }
    __syncthreads();

    int lane = tid & 31;
    int wv = tid >> 5;
    int m = lane & 15;
    int half = lane >> 4;
    int col = wv * 16 + (lane & 15);

    v8f acc[5] = {};
    for (int k0 = 0; k0 < HH; k0 += 4) {
        int ka = k0 + half * 2;
        v2f b;
        b.x = W[(size_t)ka * HH + col];
        b.y = W[(size_t)(ka + 1) * HH + col];
#pragma unroll
        for (int t = 0; t < 5; ++t) {
            v2f a;
            a.x = ldsA[(m + 16 * t) * 132 + ka];
            a.y = ldsA[(m + 16 * t) * 132 + ka + 1];
            acc[t] = __builtin_amdgcn_wmma_f32_16x16x4_f32(
                false, a, false, b, (short)0, acc[t], false, false);
        }
    }

    float bb = bias[col];
    int mbase = half * 8;
#pragma unroll
    for (int t = 0; t < 5; ++t) {
#pragma unroll
        for (int i = 0; i < 8; ++i) {
            float v = fmaxf(acc[t][i] + bb, 0.f);
            xout[(size_t)(row0 + 16 * t + mbase + i) * HH + col] = v;
        }
    }
}

// ---------------- pooling: per-graph feature sums + counts ------------------
__global__ __launch_bounds__(256) void gin_pool(
    const float* __restrict__ x, const int* __restrict__ batch,
    float* __restrict__ psum, float* __restrict__ pcnt)
{
    int lane = threadIdx.x & 31;
    int i = blockIdx.x * 8 + (threadIdx.x >> 5);
    if (i >= NN) return;
    int g = batch[i];
    float4 v = ((const float4*)(x + (size_t)i * HH))[lane];
    float* p = psum + (size_t)g * HH + lane * 4;
    atomicAdd(p + 0, v.x);
    atomicAdd(p + 1, v.y);
    atomicAdd(p + 2, v.z);
    atomicAdd(p + 3, v.w);
    if (lane == 0) atomicAdd(&pcnt[g], 1.0f);
}

// ---------------- classifier: out = (psum/cnt) @ Wc + bc --------------------
__global__ void gin_classify(const float* __restrict__ psum, const float* __restrict__ pcnt,
                             const float* __restrict__ Wc, const float* __restrict__ bc,
                             float* __restrict__ out)
{
    int t = blockIdx.x * blockDim.x + threadIdx.x;
    if (t >= GG * CC) return;
    int g = t / CC;
    int cc = t % CC;
    float inv = 1.0f / fmaxf(pcnt[g], 1.0f);
    float acc = 0.f;
    for (int j = 0; j < HH; ++j)
        acc += psum[(size_t)g * HH + j] * Wc[j * CC + cc];
    out[t] = acc * inv + bc[cc];
}

extern "C" void kernel_launch(void* const* d_in, const int* in_sizes, int n_in,
                              void* d_out, int out_size, void* d_ws, size_t ws_size,
                              hipStream_t stream)
{
    const float* x    = (const float*)d_in[0];
    const float* nc   = (const float*)d_in[1];
    const float* ec   = (const float*)d_in[2];
    const float* W1   = (const float*)d_in[3];
    const float* b1   = (const float*)d_in[4];
    const float* g1   = (const float*)d_in[5];
    const float* be1  = (const float*)d_in[6];
    const float* W2   = (const float*)d_in[7];
    const float* b2   = (const float*)d_in[8];
    const float* Wc   = (const float*)d_in[9];
    const float* bc   = (const float*)d_in[10];
    const int*  eidx  = (const int*)d_in[11];
    const int*  batch = (const int*)d_in[12];
    float* out = (float*)d_out;

    // workspace carve-up (floats)
    float* ws = (float*)d_ws;
    size_t off = 0;
    float* agg   = ws + off; off += (size_t)NN * HH;
    float* h1    = ws + off; off += (size_t)NN * HH;
    float* xA    = ws + off; off += (size_t)NN * HH;
    float* xB    = ws + off; off += (size_t)NN * HH;
    float* ssum  = ws + off; off += HH;
    float* ssq   = ws + off; off += HH;
    float* scale = ws + off; off += HH;
    float* shift = ws + off; off += HH;
    float* psum  = ws + off; off += (size_t)GG * HH;
    float* pcnt  = ws + off; off += GG;

    const int* src = eidx;        // edge_index[0]
    const int* dst = eidx + EE;   // edge_index[1]

    const float* xin = x;
    float* xcur = xA;
    float* xalt = xB;

    for (int l = 0; l < LL; ++l) {
        hipMemsetAsync(agg, 0, (size_t)NN * HH * sizeof(float), stream);
        hipMemsetAsync(ssum, 0, 2 * HH * sizeof(float), stream);  // ssum+ssq contiguous

        gin_aggregate<<<(EE + 7) / 8, 256, 0, stream>>>(xin, ec, src, dst, agg);
        gin_gemm1<<<NN / MT, 256, 0, stream>>>(agg, xin, nc,
                                               W1 + (size_t)l * HH * HH, b1 + l * HH,
                                               h1, ssum, ssq);
        gin_bnstats<<<1, HH, 0, stream>>>(ssum, ssq, g1 + l * HH, be1 + l * HH, scale, shift);
        gin_gemm2<<<NN / MT, 256, 0, stream>>>(h1, scale, shift,
                                               W2 + (size_t)l * HH * HH, b2 + l * HH, xcur);
        xin = xcur;
        float* t = xcur; xcur = xalt; xalt = t;
    }

    hipMemsetAsync(psum, 0, ((size_t)GG * HH + GG) * sizeof(float), stream); // psum+pcnt contiguous
    gin_pool<<<(NN + 7) / 8, 256, 0, stream>>>(xin, batch, psum, pcnt);
    gin_classify<<<(GG * CC + 255) / 256, 256, 0, stream>>>(psum, pcnt, Wc, bc, out);
}